// FeaturesLoss_88038239634243
// MI455X (gfx1250) — compile-verified
//
#include <hip/hip_runtime.h>
#include <hip/hip_bf16.h>

typedef __attribute__((ext_vector_type(2))) float v2f;
typedef __attribute__((ext_vector_type(8))) float v8f;

#define N_ROWS 8192
#define DIMS   64
#define MARGIN 1.0f
#define T_TILES (N_ROWS / 16)                    // 512 16-wide tile rows/cols
#define STRIP   4                                // tiles per wave along J
#define J_GROUPS (T_TILES / STRIP)               // 128
#define WAVES_PER_BLOCK 8
#define BLOCK_ROWS (T_TILES / WAVES_PER_BLOCK)   // 64
#define N_PART (J_GROUPS * BLOCK_ROWS)           // 8192 block partials

// ---------------- Kernel 1: per-row squared norms ----------------
__global__ void FL_sq_kernel(const float* __restrict__ F, float* __restrict__ sq) {
    int r = blockIdx.x * blockDim.x + threadIdx.x;
    if (r < N_ROWS) {
        const float* row = F + (size_t)r * DIMS;
        float s = 0.0f;
#pragma unroll
        for (int k = 0; k < DIMS; k += 4) {
            float4 v = *(const float4*)(row + k);
            s = fmaf(v.x, v.x, s);
            s = fmaf(v.y, v.y, s);
            s = fmaf(v.z, v.z, s);
            s = fmaf(v.w, v.w, s);
        }
        sq[r] = s;
    }
}

// Epilogue for one 16x16 C tile: C/D element (e, lane) -> row = e + 8*hi, col = lane%16
__device__ __forceinline__ void FL_epi(const v8f& c, int rowBase, int hi, int j,
                                       const float* __restrict__ sq,
                                       const int* __restrict__ cls,
                                       const float sqi[8], const int ci[8],
                                       float& pos, float& neg, unsigned& cnt)
{
    const float sqj = sq[j];
    const int   cj  = cls[j];
#pragma unroll
    for (int e = 0; e < 8; ++e) {
        const int i = rowBase + e + 8 * hi;
        if (i < j) {  // strict upper triangle (also prunes below-diagonal tiles)
            float d2 = fmaxf(sqi[e] + sqj - 2.0f * c[e], 0.0f);
            if (ci[e] == cj) {
                pos += d2;
                cnt += 1u;
            } else {
                float d = sqrtf(fmaxf(d2, 1e-12f));
                float t = fmaxf(MARGIN - d, 0.0f);
                neg = fmaf(t, t, neg);
            }
        }
    }
}

// ---------------- Kernel 2: fused WMMA Gram strip + loss epilogue ----------------
// Wave w of block (bx, by) computes a 16x64 strip of G = F F^T:
//   rows  I = by*8 + w  (16 rows), cols J = bx*64 .. bx*64+63 (4 WMMA tiles).
// Per K-step: 1 A-frag load + 4 B-frag loads feed 4 v_wmma_f32_16x16x4_f32.
__global__ void __launch_bounds__(256) FL_tile_kernel(
    const float* __restrict__ F, const int* __restrict__ cls,
    const float* __restrict__ sq,
    float* __restrict__ posP, float* __restrict__ negP, unsigned* __restrict__ cntP)
{
    const int Jg   = blockIdx.x;                      // 0..127
    const int w    = threadIdx.x >> 5;
    const int It   = blockIdx.y * WAVES_PER_BLOCK + w;
    const int lane = threadIdx.x & 31;

    float pos = 0.0f, neg = 0.0f;
    unsigned cnt = 0;

    // Strip has at least one upper-triangular tile iff It <= 4*Jg+3 (wave-uniform)
    if (It <= STRIP * Jg + (STRIP - 1)) {
        const int rowBase = It * 16;
        const int colBase = Jg * (16 * STRIP);
        const int lo = lane & 15;
        const int hi = lane >> 4;

        // f32 16x16x4 fragment layout: lanes 0-15 -> K={k,k+1}, lanes 16-31 ->
        // K={k+2,k+3}; M (A) / N (B) = lane%16. B = F^T, so A and B fragments
        // use the same coalesced float2 pattern.
        const float* Ap = F + (size_t)(rowBase + lo) * DIMS + 2 * hi;
        const float* B0 = F + (size_t)(colBase +  0 + lo) * DIMS + 2 * hi;
        const float* B1 = F + (size_t)(colBase + 16 + lo) * DIMS + 2 * hi;
        const float* B2 = F + (size_t)(colBase + 32 + lo) * DIMS + 2 * hi;
        const float* B3 = F + (size_t)(colBase + 48 + lo) * DIMS + 2 * hi;

        v8f c0 = {0.f,0.f,0.f,0.f,0.f,0.f,0.f,0.f};
        v8f c1 = c0, c2 = c0, c3 = c0;
#pragma unroll
        for (int k = 0; k < DIMS; k += 4) {
            v2f a  = *(const v2f*)(Ap + k);
            v2f b0 = *(const v2f*)(B0 + k);
            v2f b1 = *(const v2f*)(B1 + k);
            v2f b2 = *(const v2f*)(B2 + k);
            v2f b3 = *(const v2f*)(B3 + k);
            c0 = __builtin_amdgcn_wmma_f32_16x16x4_f32(false, a, false, b0, (short)0, c0, false, false);
            c1 = __builtin_amdgcn_wmma_f32_16x16x4_f32(false, a, false, b1, (short)0, c1, false, false);
            c2 = __builtin_amdgcn_wmma_f32_16x16x4_f32(false, a, false, b2, (short)0, c2, false, false);
            c3 = __builtin_amdgcn_wmma_f32_16x16x4_f32(false, a, false, b3, (short)0, c3, false, false);
        }

        // Row-side values loaded once, reused across the 4 column tiles
        float sqi[8]; int ci[8];
#pragma unroll
        for (int e = 0; e < 8; ++e) {
            const int i = rowBase + e + 8 * hi;
            sqi[e] = sq[i];
            ci[e]  = cls[i];
        }
        FL_epi(c0, rowBase, hi, colBase +  0 + lo, sq, cls, sqi, ci, pos, neg, cnt);
        FL_epi(c1, rowBase, hi, colBase + 16 + lo, sq, cls, sqi, ci, pos, neg, cnt);
        FL_epi(c2, rowBase, hi, colBase + 32 + lo, sq, cls, sqi, ci, pos, neg, cnt);
        FL_epi(c3, rowBase, hi, colBase + 48 + lo, sq, cls, sqi, ci, pos, neg, cnt);
    }

    // wave32 shuffle reduction (deterministic order)
#pragma unroll
    for (int off = 16; off > 0; off >>= 1) {
        pos += __shfl_down(pos, off, 32);
        neg += __shfl_down(neg, off, 32);
        cnt += __shfl_down(cnt, off, 32);
    }

    __shared__ float    sp[WAVES_PER_BLOCK];
    __shared__ float    sn[WAVES_PER_BLOCK];
    __shared__ unsigned sc[WAVES_PER_BLOCK];
    if (lane == 0) { sp[w] = pos; sn[w] = neg; sc[w] = cnt; }
    __syncthreads();

    if (threadIdx.x == 0) {
        float P = 0.0f, Ng = 0.0f; unsigned C = 0;
#pragma unroll
        for (int q = 0; q < WAVES_PER_BLOCK; ++q) { P += sp[q]; Ng += sn[q]; C += sc[q]; }
        const int bid = blockIdx.y * gridDim.x + blockIdx.x;
        posP[bid] = P; negP[bid] = Ng; cntP[bid] = C;
    }
}

// ---------------- Kernel 3: deterministic final reduction ----------------
__global__ void FL_finalize_kernel(const float* __restrict__ posP,
                                   const float* __restrict__ negP,
                                   const unsigned* __restrict__ cntP,
                                   float* __restrict__ out)
{
    __shared__ float    sp[256];
    __shared__ float    sn[256];
    __shared__ unsigned sc[256];
    float P = 0.0f, Ng = 0.0f; unsigned C = 0;
    for (int t = threadIdx.x; t < N_PART; t += 256) {
        P += posP[t]; Ng += negP[t]; C += cntP[t];
    }
    sp[threadIdx.x] = P; sn[threadIdx.x] = Ng; sc[threadIdx.x] = C;
    __syncthreads();
    for (int s = 128; s > 0; s >>= 1) {
        if (threadIdx.x < s) {
            sp[threadIdx.x] += sp[threadIdx.x + s];
            sn[threadIdx.x] += sn[threadIdx.x + s];
            sc[threadIdx.x] += sc[threadIdx.x + s];
        }
        __syncthreads();
    }
    if (threadIdx.x == 0) {
        const unsigned pcnt  = sc[0];
        const unsigned total = (unsigned)(((unsigned long long)N_ROWS * (N_ROWS - 1)) / 2);
        float pos_num = (float)(N_ROWS + pcnt);           // n + #pos pairs
        unsigned negc = total - pcnt;
        if (negc < 1u) negc = 1u;
        out[0] = 0.5f * (sp[0] / pos_num + sn[0] / (float)negc);
    }
}

extern "C" void kernel_launch(void* const* d_in, const int* in_sizes, int n_in,
                              void* d_out, int out_size, void* d_ws, size_t ws_size,
                              hipStream_t stream) {
    const float* F   = (const float*)d_in[0];   // feats [8192,64] f32
    const int*   cls = (const int*)d_in[1];     // class_idxs [8192]

    // workspace layout (floats): posP[8192] | negP[8192] | cntP[8192] | sq[8192]
    float*    ws   = (float*)d_ws;
    float*    posP = ws;
    float*    negP = ws + N_PART;
    unsigned* cntP = (unsigned*)(ws + 2 * N_PART);
    float*    sqp  = ws + 3 * N_PART;

    FL_sq_kernel<<<N_ROWS / 256, 256, 0, stream>>>(F, sqp);
    FL_tile_kernel<<<dim3(J_GROUPS, BLOCK_ROWS), 256, 0, stream>>>(F, cls, sqp, posP, negP, cntP);
    FL_finalize_kernel<<<1, 256, 0, stream>>>(posP, negP, cntP, (float*)d_out);
}